// ProposalGenerator_29231547416840
// MI455X (gfx1250) — compile-verified
//
#include <hip/hip_runtime.h>
#include <hip/hip_bf16.h>
#include <stdint.h>

#define Bz 8
#define Nz 243000
#define PRE 6000
#define PRE_PAD 6016          // multiple of 32
#define POST 1000
#define WORDS 188             // PRE_PAD / 32
#define CAND 8192
#define NMS_TH 0.7f
#define MIN_SZ 16.0f

typedef unsigned int u32;
typedef unsigned long long u64;

typedef unsigned int v4u __attribute__((ext_vector_type(4)));
typedef int v8i __attribute__((ext_vector_type(8)));
typedef int v4i __attribute__((ext_vector_type(4)));

#if defined(__HIP_DEVICE_COMPILE__) && __has_builtin(__builtin_amdgcn_ballot_w32)
#define BALLOT32(p) ((u32)__builtin_amdgcn_ballot_w32(p))
#else
#define BALLOT32(p) ((u32)__ballot(p))
#endif

#if defined(__HIP_DEVICE_COMPILE__)
#if __has_builtin(__builtin_amdgcn_tensor_load_to_lds) && __has_builtin(__builtin_amdgcn_s_wait_tensorcnt)
#define HAVE_TDM 1
#endif
#endif

// Issue a 1-D contiguous TDM tile load of `ndw` dwords (<= 65535) from global
// memory into workgroup LDS at byte offset `ldsOff`. Caller waits TENSORcnt.
__device__ __forceinline__ void tdm_load_1d_dwords(const void* gsrc, u32 ldsOff, u32 ndw) {
#if defined(HAVE_TDM)
  unsigned long long ga = (unsigned long long)(uintptr_t)gsrc;
  // D# group 0: count=1 | lds_addr | global_addr[31:0] | global_addr[56:32] | type=2
  v4u g0 = {1u, ldsOff, (u32)ga, ((u32)(ga >> 32) & 0x01FFFFFFu) | (2u << 30)};
  // D# group 1: data_size=4B; tensor_dim0 = tile_dim0 = ndw; tensor_dim1 = tile_dim1 = 1
  int w0 = (int)(2u << 16);                    // workgroup_mask=0, data_size=2 (4B)
  int w1 = (int)((ndw & 0xFFFFu) << 16);       // tensor_dim0[15:0]
  int w2 = (int)((ndw >> 16) | (1u << 16));    // tensor_dim0[31:16], tensor_dim1[15:0]=1
  int w3 = (int)(ndw << 16);                   // tensor_dim1[31:16]=0, tile_dim0=ndw
  int w4 = 1;                                  // tile_dim1=1, tile_dim2=0
  int w5 = (int)ndw;                           // tensor_dim0_stride low 32
  v8i g1 = {w0, w1, w2, w3, w4, w5, 0, 0};
  v4i g2 = {0, 0, 0, 0};
  v4i g3 = {0, 0, 0, 0};
#if __clang_major__ >= 23
  v8i g4 = {0, 0, 0, 0, 0, 0, 0, 0};
  __builtin_amdgcn_tensor_load_to_lds(g0, g1, g2, g3, g4, 0);
#else
  __builtin_amdgcn_tensor_load_to_lds(g0, g1, g2, g3, 0);
#endif
#else
  (void)gsrc; (void)ldsOff; (void)ndw;
#endif
}

__device__ __forceinline__ u32 orderedKey(float f) {
  // order-preserving float -> uint (larger float => larger uint)
  u32 u = __float_as_uint(f);
  return (u & 0x80000000u) ? ~u : (u | 0x80000000u);
}

// ---------------------------------------------------------------- NMS kernel
// (defined first so the disasm snippet shows the TDM path)
// Sequential greedy NMS, one 256-thread block per batch.
// Box tile (4 x 6016 f32 = 96,256 B) is DMA'd into LDS with the TDM
// (tensor_load_to_lds + s_wait_tensorcnt). Suppression mask lives in LDS;
// per accepted box each wave handles every 8th 32-column word (up to 24) via ballot_w32.
__global__ void __launch_bounds__(256)
nms_kernel(const float* __restrict__ topk, const u32* __restrict__ aliveW,
           float* __restrict__ outB, float* __restrict__ outV) {
  int b = blockIdx.x;
  int tid = threadIdx.x;
  extern __shared__ float lds[];             // no static LDS in this kernel -> dyn base = 0
  float* X1 = lds;
  float* Y1 = lds + PRE_PAD;
  float* X2 = lds + 2 * PRE_PAD;
  float* Y2 = lds + 3 * PRE_PAD;
  u32* sup = (u32*)(lds + 4 * PRE_PAD);
  const float* src = topk + (size_t)b * 4 * PRE_PAD;

#if defined(HAVE_TDM)
  if (tid < 32) {  // wave 0 issues the single TDM op (EXEC-independent, per-wave issue)
    tdm_load_1d_dwords(src, 0u, 4u * PRE_PAD);
    __builtin_amdgcn_s_wait_tensorcnt(0);
  }
#else
  for (int t = tid; t < 4 * PRE_PAD; t += 256) lds[t] = src[t];
#endif
  for (int w = tid; w < WORDS; w += 256) sup[w] = ~aliveW[b * WORDS + w];
  __syncthreads();

  int lane = tid & 31;
  int wv = tid >> 5;      // 8 waves
  int i = 0, cnt = 0, cwIdx = -1;
  u32 cw = 0;
  while (i < PRE) {
    int w = i >> 5;
    if (w != cwIdx) { cwIdx = w; cw = sup[w]; }   // no writes since last barrier -> safe
    if ((cw >> (i & 31)) & 1u) { ++i; continue; }
    // ---- take box i (uniform decision across block) ----
    float bx1 = X1[i], by1 = Y1[i], bx2 = X2[i], by2 = Y2[i];
    float ba = (bx2 - bx1) * (by2 - by1);
    for (int s = 0; s < 24; ++s) {
      int word = wv + (s << 3);
      if (word >= WORDS) break;                 // wave-uniform exit
      int c = (word << 5) + lane;               // 32 consecutive cols per wave-step
      float cx1 = X1[c], cy1 = Y1[c], cx2 = X2[c], cy2 = Y2[c];
      float ca = (cx2 - cx1) * (cy2 - cy1);
      float ix = fminf(bx2, cx2) - fmaxf(bx1, cx1);
      float iy = fminf(by2, cy2) - fmaxf(by1, cy1);
      float inter = fmaxf(ix, 0.f) * fmaxf(iy, 0.f);
      float uni = ba + ca - inter;
      bool hit = inter > NMS_TH * uni;          // == (iou > 0.7) without division
      u32 m = BALLOT32(hit);
      if (lane == 0 && m) sup[word] |= m;       // each word touched by exactly one wave-step
    }
    if (tid < 4) outB[((size_t)(b * POST + cnt)) * 4 + tid] = lds[tid * PRE_PAD + i];
    if (tid == 0) outV[b * POST + cnt] = 1.0f;
    ++cnt;
    __syncthreads();                            // publish suppression ORs
    cw = sup[w];                                // refresh current word
    ++i;
    if (cnt >= POST) break;
  }
  // zero-pad remaining slots (matches reference: boxes*ok, ok=false)
  for (int t = cnt + tid; t < POST; t += 256) {
    outV[b * POST + t] = 0.f;
    float* o = outB + ((size_t)(b * POST + t)) * 4;
    o[0] = 0.f; o[1] = 0.f; o[2] = 0.f; o[3] = 0.f;
  }
}

// ---------------------------------------------------------------- kernel 1
// Decode anchors+deltas -> clipped proposals; masked sigmoid score -> key.
__global__ void decode_kernel(const int* __restrict__ img,
                              const float* __restrict__ anch,
                              const float* __restrict__ logit,
                              const float* __restrict__ delt,
                              float* __restrict__ props,
                              u32* __restrict__ keys) {
  int i = blockIdx.x * blockDim.x + threadIdx.x;
  if (i >= Bz * Nz) return;
  int b = i / Nz;
  const float* a = anch + (size_t)i * 4;
  const float* d = delt + (size_t)i * 4;
  float ax1 = a[0], ay1 = a[1], ax2 = a[2], ay2 = a[3];
  float aw = ax2 - ax1, ah = ay2 - ay1;
  float acx = ax1 + 0.5f * aw, acy = ay1 + 0.5f * ah;
  float px = acx + d[0] * aw;
  float py = acy + d[1] * ah;
  float pw = aw * expf(fminf(fmaxf(d[2], -10.f), 10.f));
  float ph = ah * expf(fminf(fmaxf(d[3], -10.f), 10.f));
  float H = (float)img[b * 2 + 0], W = (float)img[b * 2 + 1];
  float x1 = fminf(fmaxf(px - 0.5f * pw, 0.f), W - 1.f);
  float y1 = fminf(fmaxf(py - 0.5f * ph, 0.f), H - 1.f);
  float x2 = fminf(fmaxf(px + 0.5f * pw, 0.f), W - 1.f);
  float y2 = fminf(fmaxf(py + 0.5f * ph, 0.f), H - 1.f);
  bool valid = (x2 - x1 >= MIN_SZ) && (y2 - y1 >= MIN_SZ);
  float fg = 1.f / (1.f + expf(-logit[i]));
  float masked = valid ? fg : -1.f;
  float* p = props + (size_t)i * 4;
  p[0] = x1; p[1] = y1; p[2] = x2; p[3] = y2;
  keys[i] = orderedKey(masked);
}

// ---------------------------------------------------------------- kernel 2
// Per-batch MSB-first radix select: exact key value of the PRE-th largest.
// Also zero-initializes the per-batch scratch used by later kernels.
__global__ void __launch_bounds__(1024)
select_kernel(const u32* __restrict__ keys, u32* __restrict__ thresholds,
              u32* __restrict__ counts, u64* __restrict__ cand,
              u32* __restrict__ aliveW) {
  int b = blockIdx.x;
  int tid = threadIdx.x;
  const u32* kb = keys + (size_t)b * Nz;
  for (int t = tid; t < CAND; t += 1024) cand[(size_t)b * CAND + t] = 0ull;
  for (int t = tid; t < WORDS; t += 1024) aliveW[b * WORDS + t] = 0u;
  if (tid == 0) counts[b] = 0u;

  __shared__ u32 hist[256];
  __shared__ u32 sh_sel, sh_k;
  u32 prefix = 0, pmask = 0, k = PRE;
  for (int pass = 3; pass >= 0; --pass) {
    if (tid < 256) hist[tid] = 0u;
    __syncthreads();
    int shp = pass * 8;
    for (int idx = tid; idx < Nz; idx += 1024) {
      if (idx + 8192 < Nz) __builtin_prefetch(kb + idx + 8192, 0, 0);  // global_prefetch_b8
      u32 key = kb[idx];
      if ((key & pmask) == prefix) atomicAdd(&hist[(key >> shp) & 255u], 1u);
    }
    __syncthreads();
    if (tid == 0) {
      u32 cum = 0;
      int d = 255;
      for (; d > 0; --d) {
        u32 h = hist[d];
        if (cum + h >= k) break;
        cum += h;
      }
      sh_sel = (u32)d;
      sh_k = k - cum;
    }
    __syncthreads();
    prefix |= sh_sel << shp;
    pmask |= 255u << shp;
    k = sh_k;
    __syncthreads();
  }
  if (tid == 0) thresholds[b] = prefix;
}

// ---------------------------------------------------------------- kernel 3
// Compact all candidates with key >= threshold (count in [6000, 6000+ties]).
__global__ void compact_kernel(const u32* __restrict__ keys,
                               const u32* __restrict__ thresholds,
                               u32* __restrict__ counts, u64* __restrict__ cand) {
  int i = blockIdx.x * blockDim.x + threadIdx.x;
  if (i >= Bz * Nz) return;
  int b = i / Nz;
  u32 key = keys[i];
  if (key >= thresholds[b]) {
    u32 pos = atomicAdd(&counts[b], 1u);
    if (pos < CAND) {
      u32 n = (u32)(i - b * Nz);
      cand[(size_t)b * CAND + pos] = ((u64)key << 32) | (u32)(~n);  // ~n: desc sort => idx asc ties
    }
  }
}

// ---------------------------------------------------------------- kernel 4
// Per-batch bitonic sort (descending) of 8192 u64 pairs in 64KB LDS (loaded
// via TDM tile DMA), then gather top-PRE boxes into SoA [b][comp][r] layout
// + alive bitmask.
__global__ void __launch_bounds__(1024)
sort_kernel(const u64* __restrict__ cand, const float* __restrict__ props,
            float* __restrict__ topk, u32* __restrict__ aliveW) {
  int b = blockIdx.x;
  int tid = threadIdx.x;
  extern __shared__ u64 s64[];               // no static LDS in this kernel -> dyn base = 0
#if defined(HAVE_TDM)
  if (tid < 32) {  // wave 0 issues the TDM op: 8192 u64 = 16384 dwords = 64KB tile
    tdm_load_1d_dwords(cand + (size_t)b * CAND, 0u, CAND * 2u);
    __builtin_amdgcn_s_wait_tensorcnt(0);
  }
#else
  for (int t = tid; t < CAND; t += 1024) s64[t] = cand[(size_t)b * CAND + t];
#endif
  __syncthreads();
  for (int k2 = 2; k2 <= CAND; k2 <<= 1) {
    for (int j = k2 >> 1; j > 0; j >>= 1) {
      for (int t = tid; t < CAND; t += 1024) {
        int ixj = t ^ j;
        if (ixj > t) {
          u64 a = s64[t], c = s64[ixj];
          bool descBlock = ((t & k2) == 0);
          if (descBlock ? (a < c) : (a > c)) { s64[t] = c; s64[ixj] = a; }
        }
      }
      __syncthreads();
    }
  }
  float* tb = topk + (size_t)b * 4 * PRE_PAD;
  for (int r = tid; r < PRE_PAD; r += 1024) {
    float x1 = 0, y1 = 0, x2 = 0, y2 = 0;
    bool alive = false;
    if (r < PRE) {
      u64 e = s64[r];
      u32 key = (u32)(e >> 32);
      if (key != 0u) {                      // 0 == padding sentinel
        u32 idx = ~((u32)e);
        if (idx < Nz) {
          const float* p = props + (((size_t)b * Nz) + idx) * 4;
          x1 = p[0]; y1 = p[1]; x2 = p[2]; y2 = p[3];
        }
        alive = key > 0x80000000u;          // masked score > 0.0
      }
    }
    tb[0 * PRE_PAD + r] = x1;
    tb[1 * PRE_PAD + r] = y1;
    tb[2 * PRE_PAD + r] = x2;
    tb[3 * PRE_PAD + r] = y2;
    if (alive) atomicOr(&aliveW[b * WORDS + (r >> 5)], 1u << (r & 31));
  }
}

// ----------------------------------------------------------------
extern "C" void kernel_launch(void* const* d_in, const int* in_sizes, int n_in,
                              void* d_out, int out_size, void* d_ws, size_t ws_size,
                              hipStream_t stream) {
  (void)in_sizes; (void)n_in; (void)out_size; (void)ws_size;
  const int* img = (const int*)d_in[0];
  const float* anch = (const float*)d_in[1];
  const float* logit = (const float*)d_in[2];
  const float* delt = (const float*)d_in[3];

  char* ws = (char*)d_ws;
  float* props = (float*)ws;      ws += (size_t)Bz * Nz * 4 * sizeof(float);   // 31.1 MB
  u32* keys = (u32*)ws;           ws += (size_t)Bz * Nz * sizeof(u32);         // 7.78 MB
  u64* cand = (u64*)ws;           ws += (size_t)Bz * CAND * sizeof(u64);       // 0.52 MB (8B aligned)
  float* topk = (float*)ws;       ws += (size_t)Bz * 4 * PRE_PAD * sizeof(float);
  u32* thresholds = (u32*)ws;     ws += Bz * sizeof(u32);
  u32* counts = (u32*)ws;         ws += Bz * sizeof(u32);
  u32* aliveW = (u32*)ws;         ws += Bz * WORDS * sizeof(u32);

  float* outB = (float*)d_out;                 // (B, 1000, 4)
  float* outV = outB + (size_t)Bz * POST * 4;  // (B, 1000)

  // Allow >64KB dynamic LDS (CDNA5 WGP has 320KB)
  hipFuncSetAttribute(reinterpret_cast<const void*>(sort_kernel),
                      hipFuncAttributeMaxDynamicSharedMemorySize, CAND * (int)sizeof(u64));
  hipFuncSetAttribute(reinterpret_cast<const void*>(nms_kernel),
                      hipFuncAttributeMaxDynamicSharedMemorySize,
                      4 * PRE_PAD * (int)sizeof(float) + WORDS * (int)sizeof(u32));

  int total = Bz * Nz;
  decode_kernel<<<(total + 255) / 256, 256, 0, stream>>>(img, anch, logit, delt, props, keys);
  select_kernel<<<Bz, 1024, 0, stream>>>(keys, thresholds, counts, cand, aliveW);
  compact_kernel<<<(total + 255) / 256, 256, 0, stream>>>(keys, thresholds, counts, cand);
  sort_kernel<<<Bz, 1024, CAND * sizeof(u64), stream>>>(cand, props, topk, aliveW);
  nms_kernel<<<Bz, 256, 4 * PRE_PAD * sizeof(float) + WORDS * sizeof(u32), stream>>>(
      topk, aliveW, outB, outV);
}